// PCSamplingProcessor_31610959298724
// MI455X (gfx1250) — compile-verified
//
#include <hip/hip_runtime.h>
#include <hip/hip_bf16.h>
#include <math.h>

#define B_BATCH 8
#define NPTS    8192
#define NSAMP   1024
#define KNN     24
#define FPS_THREADS 1024
#define PPT     (NPTS / FPS_THREADS)

#define KNN_BLOCK       128
#define TILES_PER_BLOCK 4

typedef __attribute__((ext_vector_type(2))) float v2f;
typedef __attribute__((ext_vector_type(8))) float v8f;
typedef unsigned long long u64;
typedef unsigned int u32;

#if defined(__HIP_DEVICE_COMPILE__) && __has_builtin(__builtin_amdgcn_tensor_load_to_lds)
#define HAVE_TDM 1
typedef __attribute__((ext_vector_type(4))) unsigned tdm_v4u;
typedef __attribute__((ext_vector_type(8))) int      tdm_v8i;
typedef __attribute__((ext_vector_type(4))) int      tdm_v4i;
#else
#define HAVE_TDM 0
#endif

// ---------------------------------------------------------------------------
// Phase 1: farthest point sampling. One workgroup per batch; all 8192 points
// staged in LDS; per-thread running min-distance in registers. Argmax uses a
// packed u64 key: (dist_bits << 32) | ~index  -> 64-bit max gives the largest
// distance with smallest index (jnp.argmax first-match semantics), fully
// branchless. Writes new_xyz directly; also zeroes the global accumulators.
// ---------------------------------------------------------------------------
__global__ __launch_bounds__(FPS_THREADS)
void fps_kernel(const float* __restrict__ xyz, float* __restrict__ new_xyz,
                double* __restrict__ acc) {
  extern __shared__ __align__(16) unsigned char smem_raw[];
  float* sx   = (float*)smem_raw;
  float* sy   = sx + NPTS;
  float* sz   = sy + NPTS;
  u64*   rk   = (u64*)(sz + NPTS);     // 32 keys (8B aligned)
  int*   sfar = (int*)(rk + 32);       // 1 int

  const int b = blockIdx.x;
  const int t = threadIdx.x;
  if (b == 0 && t == 0) { acc[0] = 0.0; acc[1] = 0.0; }

  const float* src = xyz + (size_t)b * NPTS * 3;
  float dist[PPT];
#pragma unroll
  for (int k = 0; k < PPT; ++k) {
    int p = k * FPS_THREADS + t;
    sx[p] = src[p * 3 + 0];
    sy[p] = src[p * 3 + 1];
    sz[p] = src[p * 3 + 2];
    dist[k] = 1e10f;
  }
  __syncthreads();

  float* out = new_xyz + (size_t)b * NSAMP * 3;
  const int wave = t >> 5, lane = t & 31;
  int far = 0;
  for (int s = 0; s < NSAMP; ++s) {
    float cx = sx[far], cy = sy[far], cz = sz[far];
    if (t == 0) { out[s*3+0] = cx; out[s*3+1] = cy; out[s*3+2] = cz; }
    u64 key = 0ull;
#pragma unroll
    for (int k = 0; k < PPT; ++k) {
      int p = k * FPS_THREADS + t;
      float dx = sx[p]-cx, dy = sy[p]-cy, dz = sz[p]-cz;
      float d  = dx*dx + dy*dy + dz*dz;
      float nd = fminf(dist[k], d);
      dist[k] = nd;
      u64 ck = ((u64)__float_as_uint(nd) << 32) | (u32)(~p);
      key = (ck > key) ? ck : key;
    }
#pragma unroll
    for (int off = 16; off > 0; off >>= 1) {
      u64 ok = __shfl_down(key, off);
      key = (ok > key) ? ok : key;
    }
    if (lane == 0) rk[wave] = key;
    __syncthreads();
    if (t < 32) {
      key = rk[t];
#pragma unroll
      for (int off = 16; off > 0; off >>= 1) {
        u64 ok = __shfl_down(key, off);
        key = (ok > key) ? ok : key;
      }
      if (t == 0) *sfar = (int)(~(u32)key);
    }
    __syncthreads();
    far = *sfar;
    __syncthreads();
  }
}

// ---------------------------------------------------------------------------
// Phase 2: KNN via V_WMMA_F32_16X16X4_F32 (one wave per 16 sampled points).
// All 4 waves of a block scan the same batch's 8192 candidates, so the whole
// batch (96KB) is staged into LDS once per block -- via the Tensor Data Mover
// (TENSOR_LOAD_TO_LDS, one 1-row D# descriptor + s_wait_tensorcnt) when the
// toolchain exposes the builtin, otherwise a cooperative b128 copy.
// K-slot packing: A = candidates [x, y | z, |c|^2], B = sampled
// [-2sx, -2sy | -2sz, 1.0], C = |s|^2 -> D = full squared distance in one
// WMMA. Per-lane sorted top-24 of packed u64 keys (dist_bits<<32 | idx) via a
// branchless v_min_u64/v_max_u64 bubble; lane n / n+16 lists merged in LDS.
// ---------------------------------------------------------------------------
__global__ __launch_bounds__(KNN_BLOCK)
void knn_kernel(const float* __restrict__ xyz, const float* __restrict__ new_xyz,
                float* __restrict__ knn_out, double* __restrict__ acc) {
  extern __shared__ __align__(16) unsigned char kshmem[];
  float*  cand = (float*)kshmem;                                   // [NPTS*3]
  u64*    mk   = (u64*)(kshmem + (size_t)NPTS * 3 * sizeof(float));// [128][KNN]
  double* rsum = (double*)(mk + KNN_BLOCK * KNN);                  // [4]
  double* rsq  = rsum + TILES_PER_BLOCK;                           // [4]

  const int tid  = threadIdx.x;
  const int wave = tid >> 5;
  const int lane = tid & 31;
  const int tile = blockIdx.x * TILES_PER_BLOCK + wave;
  const int b    = tile >> 6;            // all 4 tiles in a block: same batch
  const int s0   = (tile & 63) << 4;
  const int half = (lane >= 16) ? 1 : 0;
  const int l16  = lane & 15;
  const int moff = half ? 8 : 0;

  const float* xb = xyz + (size_t)b * NPTS * 3;

  // ---- Stage the batch's candidate points into LDS ----
#if HAVE_TDM
  if (wave == 0) {
    unsigned lds_off = (unsigned)(size_t)(void*)cand;   // LDS aperture: addr[31:0]
    unsigned long long ga = (unsigned long long)(size_t)xb;
    tdm_v4u g0;
    g0[0] = 1u;                                         // count=1 (valid), user mode
    g0[1] = lds_off;                                    // lds_addr (bytes)
    g0[2] = (unsigned)(ga & 0xFFFFFFFFull);             // global_addr[31:0]
    g0[3] = (unsigned)((ga >> 32) & 0x01FFFFFFull)      // global_addr[56:32]
          | (2u << 30);                                 // type=2 ("image")
    tdm_v8i g1;
    g1[0] = (int)(2u << 16);                            // data_size=4B, mask=0
    g1[1] = (int)((u32)(NPTS * 3) << 16);               // tensor_dim0[15:0]
    g1[2] = (int)(1u << 16);                            // dim0 hi=0, tensor_dim1=1
    g1[3] = (int)((u32)(NPTS * 3) << 16);               // tile_dim0 = 24576
    g1[4] = 0;                                          // tile_dim1/2 unused
    g1[5] = (int)(u32)(NPTS * 3);                       // tensor_dim0_stride
    g1[6] = 0;
    g1[7] = 0;
    tdm_v4i g2 = (tdm_v4i)0;
    tdm_v4i g3 = (tdm_v4i)0;
#if defined(__clang_major__) && (__clang_major__ >= 23)
    tdm_v8i g4 = (tdm_v8i)0;
    __builtin_amdgcn_tensor_load_to_lds(g0, g1, g2, g3, g4, 0);
#else
    __builtin_amdgcn_tensor_load_to_lds(g0, g1, g2, g3, 0);
#endif
    __builtin_amdgcn_s_wait_tensorcnt(0);
  }
  __syncthreads();
#else
  for (int i = tid; i < (NPTS * 3) / 4; i += KNN_BLOCK)
    ((float4*)cand)[i] = ((const float4*)xb)[i];
  __syncthreads();
#endif

  const int sn = s0 + l16;
  const float* sp = new_xyz + ((size_t)b * NSAMP + sn) * 3;
  const float sxc = sp[0], syc = sp[1], szc = sp[2];
  const float sn2 = sxc*sxc + syc*syc + szc*szc;

  // B fragment (4x16): VGPR0 = K0/K2 halves, VGPR1 = K1/K3 halves.
  v2f bf;
  bf[0] = half ? (-2.0f * szc) : (-2.0f * sxc);
  bf[1] = half ? 1.0f          : (-2.0f * syc);

  v8f cf;
#pragma unroll
  for (int j = 0; j < 8; ++j) cf[j] = sn2;

  u64 hk[KNN];
#pragma unroll
  for (int i = 0; i < KNN; ++i) hk[i] = ~0ull;

  for (int c0 = 0; c0 < NPTS; c0 += 16) {
    const int ci0 = c0 + l16;
    const float cx = cand[ci0*3+0], cy = cand[ci0*3+1], cz = cand[ci0*3+2];
    const float cn2 = cx*cx + cy*cy + cz*cz;

    // A fragment (16x4): lanes<16 {K0,K1}={x,y}; lanes>=16 {K2,K3}={z,|c|^2}
    v2f af;
    af[0] = half ? cz  : cx;
    af[1] = half ? cn2 : cy;

    v8f df = __builtin_amdgcn_wmma_f32_16x16x4_f32(
        false, af, false, bf, (short)0, cf, false, false);

#pragma unroll
    for (int j = 0; j < 8; ++j) {
      float dj = fmaxf(df[j], 0.0f);                 // keep bits order-monotone
      u64 key = ((u64)__float_as_uint(dj) << 32) | (u32)(c0 + moff + j);
      if (key < hk[KNN-1]) {
#pragma unroll
        for (int i = 0; i < KNN; ++i) {
          u64 h  = hk[i];
          bool sw = key < h;
          hk[i] = sw ? key : h;                      // min
          key   = sw ? h   : key;                    // max bubbles up
        }
      }
    }
  }

  // Dump per-lane sorted key lists, merge lane n with lane n+16.
#pragma unroll
  for (int i = 0; i < KNN; ++i) mk[tid * KNN + i] = hk[i];
  __syncthreads();

  float lsum = 0.0f, lsq = 0.0f;
  if (!half) {
    const u64* ka = mk + (size_t)tid * KNN;
    const u64* kb = mk + (size_t)(tid + 16) * KNN;
    int ia = 0, ib = 0;
    float* outp = knn_out + (((size_t)b * NSAMP + sn) * KNN) * 6;
    for (int k = 0; k < KNN; ++k) {
      u64 va = ka[ia], vb = kb[ib];
      bool ta = va <= vb;
      u64 v = ta ? va : vb;
      if (ta) ++ia; else ++ib;
      int ci = (int)(u32)(v & 0xFFFFFFFFull);
      float gx = cand[ci*3+0], gy = cand[ci*3+1], gz = cand[ci*3+2];
      float dx = gx - sxc, dy = gy - syc, dz = gz - szc;
      outp[k*6+0] = dx;  outp[k*6+1] = dy;  outp[k*6+2] = dz;
      outp[k*6+3] = sxc; outp[k*6+4] = syc; outp[k*6+5] = szc;
      lsum += dx + dy + dz;
      lsq  += dx*dx + dy*dy + dz*dz;
    }
  }
#pragma unroll
  for (int off = 16; off > 0; off >>= 1) {
    lsum += __shfl_down(lsum, off);
    lsq  += __shfl_down(lsq,  off);
  }
  if (lane == 0) { rsum[wave] = (double)lsum; rsq[wave] = (double)lsq; }
  __syncthreads();
  if (tid == 0) {
    double s = 0.0, q = 0.0;
    for (int w = 0; w < TILES_PER_BLOCK; ++w) { s += rsum[w]; q += rsq[w]; }
    atomicAdd(&acc[0], s);
    atomicAdd(&acc[1], q);
  }
}

// ---------------------------------------------------------------------------
// Phase 3: global ddof=1 std from the accumulated moments; rescale diff halves.
// ---------------------------------------------------------------------------
__global__ __launch_bounds__(256)
void norm_kernel(float* __restrict__ knn, const double* __restrict__ acc) {
  const double n = (double)B_BATCH * NSAMP * KNN * 3;
  double s = acc[0], q = acc[1];
  double var = (q - s * s / n) / (n - 1.0);
  float scale = 1.0f / ((float)sqrt(var) + 1e-5f);
  int i = blockIdx.x * blockDim.x + threadIdx.x;
  if (i < B_BATCH * NSAMP * KNN) {
    float* p = knn + (size_t)i * 6;
    p[0] *= scale; p[1] *= scale; p[2] *= scale;
  }
}

extern "C" void kernel_launch(void* const* d_in, const int* in_sizes, int n_in,
                              void* d_out, int out_size, void* d_ws, size_t ws_size,
                              hipStream_t stream) {
  const float* xyz = (const float*)d_in[0];
  float* out      = (float*)d_out;
  float* new_xyz  = out;                                   // [8,1024,3]
  float* knn      = out + (size_t)B_BATCH * NSAMP * 3;     // [8,1024,24,6]
  double* acc     = (double*)d_ws;

  size_t fps_lds = (size_t)3 * NPTS * sizeof(float)
                 + 32 * sizeof(u64) + sizeof(int);

  fps_kernel<<<B_BATCH, FPS_THREADS, fps_lds, stream>>>(xyz, new_xyz, acc);

  size_t knn_lds = (size_t)NPTS * 3 * sizeof(float)        // candidate stage
                 + (size_t)KNN_BLOCK * KNN * sizeof(u64)   // merge keys
                 + 2 * TILES_PER_BLOCK * sizeof(double);   // block moments
  int knn_blocks = (B_BATCH * (NSAMP / 16)) / TILES_PER_BLOCK;  // 128
  knn_kernel<<<knn_blocks, KNN_BLOCK, knn_lds, stream>>>(xyz, new_xyz, knn, acc);

  int tot = B_BATCH * NSAMP * KNN;
  norm_kernel<<<(tot + 255) / 256, 256, 0, stream>>>(knn, acc);
}